// S4DKernel_10479720202534
// MI455X (gfx1250) — compile-verified
//
#include <hip/hip_runtime.h>
#include <hip/hip_bf16.h>
#include <math.h>

typedef float v2f __attribute__((ext_vector_type(2)));
typedef float v8f __attribute__((ext_vector_type(8)));

#define H_DIM          1024
#define N_HALF_DIM     32
#define TILE_L         256
#define WAVES_PER_BLK  8

// One wave32 per (h, 256-wide l-tile).
// out[h, l0 + 16*m + c] = 2 * Re( G @ P )[m,c]
//   G[m,n] = Ceff[n] * exp(dtA[n]*(l0+16m))   (16 x 32 complex)
//   P[n,c] = exp(dtA[n]*c)                    (32 x 16 complex)
// Real expansion -> 16x16x64 fp32 GEMM = 16 x V_WMMA_F32_16X16X4_F32.
//
// All transcendentals use the hardware TRANS ops (v_exp_f32 / v_sin_f32 /
// v_cos_f32): branch-free (EXEC stays all-1s for WMMA), and accurate wherever
// the decaying envelope exp(dtAr*l) leaves the term non-negligible.
__global__ __launch_bounds__(256) void s4d_vandermonde_wmma(
    const float* __restrict__ C_param,     // (H, 32, 2)
    const float* __restrict__ log_dt,      // (H,)
    const float* __restrict__ log_A_real,  // (H, 32)
    const float* __restrict__ A_imag,      // (H, 32)
    float* __restrict__ out,               // (H, L)
    int L)
{
    const int lane = threadIdx.x & 31;
    const int wave = threadIdx.x >> 5;
    const int tilesPerH = L / TILE_L;                         // 16
    const int tile = blockIdx.x * WAVES_PER_BLK + wave;       // uniform, no tail
    const int h    = tile / tilesPerH;
    const int l0   = (tile - h * tilesPerH) * TILE_L;

    // ---- per-lane state for mode n = lane --------------------------------
    const int n = lane;
    const int base = h * N_HALF_DIM + n;

    const float dtv = __expf(log_dt[h]);
    const float Ar  = -__expf(log_A_real[base]);
    const float Ai  = A_imag[base];
    const float c0  = C_param[2 * base + 0];
    const float c1  = C_param[2 * base + 1];

    const float dtAr = Ar * dtv;
    const float dtAi = Ai * dtv;

    // w = exp(dtA) - 1 ; Ceff = C * w / A  (divide via conj(A)/|A|^2)
    float sw, cw;
    __sincosf(dtAi, &sw, &cw);
    const float ew  = __expf(dtAr);
    const float wr  = ew * cw - 1.0f;
    const float wi  = ew * sw;
    const float inv = 1.0f / (Ar * Ar + Ai * Ai);
    const float tr  = (wr * Ar + wi * Ai) * inv;
    const float ti  = (wi * Ar - wr * Ai) * inv;
    const float Cr  = c0 * tr - c1 * ti;    // Re(Ceff)
    const float Ci  = c0 * ti + c1 * tr;    // Im(Ceff)

    // ---- fragment index decomposition ------------------------------------
    const int half = lane >> 4;   // 0: lanes 0-15, 1: lanes 16-31
    const int m    = lane & 15;   // A: row M     B: col c (N)    D: col N

    v8f acc = {};

    #pragma unroll
    for (int kk = 0; kk < 16; ++kk) {
        // This step consumes k = 4*kk + {0,1,2,3}; this lane's half handles
        // k = 4*kk + 2*half + {0,1}  -> complex mode n0 = 2*kk + half.
        const int n0 = 2 * kk + half;
        const float dAr = __shfl(dtAr, n0, 32);
        const float dAi = __shfl(dtAi, n0, 32);
        const float gcr = __shfl(Cr,   n0, 32);
        const float gci = __shfl(Ci,   n0, 32);

        // B fragment: b.x = Re P[n0, c] (k even), b.y = Im P[n0, c] (k odd)
        const float fc = (float)m;
        float sb, cb;
        __sincosf(dAi * fc, &sb, &cb);
        const float eb = __expf(dAr * fc);
        v2f b;
        b.x = eb * cb;
        b.y = eb * sb;

        // A fragment: a.x = Re G[m, n0], a.y = -Im G[m, n0]
        const float ph = (float)(l0 + 16 * m);
        float sa, ca;
        __sincosf(dAi * ph, &sa, &ca);
        const float ea = __expf(dAr * ph);
        const float er = ea * ca;
        const float ei = ea * sa;
        v2f a;
        a.x = gcr * er - gci * ei;      //  Re(Ceff * r^ph)
        a.y = -(gcr * ei + gci * er);   // -Im(Ceff * r^ph)

        // D = A(16x4) * B(4x16) + C, fp32
        acc = __builtin_amdgcn_wmma_f32_16x16x4_f32(
            /*neg_a=*/false, a, /*neg_b=*/false, b,
            /*c_mod=*/(short)0, acc, /*reuse_a=*/false, /*reuse_b=*/false);
    }

    // ---- store: D layout -> lane holds (M = v + 8*half, N = m) -----------
    float* op = out + (size_t)h * (size_t)L + l0;
    #pragma unroll
    for (int v = 0; v < 8; ++v) {
        const int row = v + 8 * half;
        op[row * 16 + m] = 2.0f * acc[v];
    }
}

extern "C" void kernel_launch(void* const* d_in, const int* in_sizes, int n_in,
                              void* d_out, int out_size, void* d_ws, size_t ws_size,
                              hipStream_t stream) {
    const float* C_param    = (const float*)d_in[0];
    const float* log_dt     = (const float*)d_in[1];
    const float* log_A_real = (const float*)d_in[2];
    const float* A_imag     = (const float*)d_in[3];
    // d_in[4] holds L on device; derive it host-side from out_size instead.
    const int L = out_size / H_DIM;                 // 4096
    float* out = (float*)d_out;

    const int tiles  = H_DIM * (L / TILE_L);        // 16384 wave-tiles
    const int blocks = tiles / WAVES_PER_BLK;       // 2048
    s4d_vandermonde_wmma<<<dim3(blocks), dim3(256), 0, stream>>>(
        C_param, log_dt, log_A_real, A_imag, out, L);
}